// Ranker_8272107012442
// MI455X (gfx1250) — compile-verified
//
#include <hip/hip_runtime.h>
#include <cstdint>

#define LOG2E 1.44269504088896340736f
#define LN2   0.69314718055994530942f

typedef float fx4 __attribute__((ext_vector_type(4)));   // native vector: OK for nontemporal builtin

#if defined(__gfx1250__) && __has_builtin(__builtin_amdgcn_tensor_load_to_lds) && __has_builtin(__builtin_amdgcn_s_wait_tensorcnt)
#define USE_TDM 1
typedef unsigned int u32x4 __attribute__((ext_vector_type(4)));
typedef int          i32x8 __attribute__((ext_vector_type(8)));
typedef int          i32x4 __attribute__((ext_vector_type(4)));
#endif

#if __has_builtin(__builtin_nontemporal_load)
#define NT_LOAD(p) __builtin_nontemporal_load(p)
#else
#define NT_LOAD(p) (*(p))
#endif

__device__ __forceinline__ float fast_exp2(float x) {
#if __has_builtin(__builtin_amdgcn_exp2f)
    return __builtin_amdgcn_exp2f(x);
#else
    return exp2f(x);
#endif
}
__device__ __forceinline__ float fast_log2(float x) {
#if __has_builtin(__builtin_amdgcn_logf)
    return __builtin_amdgcn_logf(x);
#else
    return log2f(x);
#endif
}

// One workgroup (8 wave32) per row. Streams the 50257-float row once with
// non-temporal b128 loads, accumulating sum(exp(x - predict)) and
// count(x > predict). seqs row staged to LDS via the Tensor Data Mover.
__global__ __launch_bounds__(256) void Ranker_row_kernel(
    const float* __restrict__ scores, const int* __restrict__ labels,
    const int* __restrict__ seqs, float* __restrict__ rank_out,
    float* __restrict__ loss_out, int V, int L)
{
    const int row = blockIdx.x;
    const int tid = threadIdx.x;
    const float* __restrict__ rowp = scores + (size_t)row * (size_t)V;

    __shared__ int   s_seqs[256];     // L = 200 <= 256
    __shared__ float s_sum[8];
    __shared__ int   s_cnt[8];
    __shared__ int   s_seqcnt;

    if (L > 256) L = 256;             // LDS staging bound (L is 200 in practice)

    const int   label   = labels[row];
    const float predict = rowp[label];
    const float negpk   = -predict * LOG2E;

    if (tid == 0) s_seqcnt = 0;

    // ---- stage seqs row into LDS (Tensor Data Mover on CDNA5) ----
#ifdef USE_TDM
    if (tid < 32) {  // one wave issues the DMA; EXEC-independent, TENSORcnt-tracked
        const uint64_t ga      = (uint64_t)(uintptr_t)(seqs + (size_t)row * (size_t)L);
        const uint32_t lds_off = (uint32_t)(uintptr_t)(&s_seqs[0]); // low 32b of shared addr = LDS byte offset
        u32x4 g0;
        g0[0] = 1u;                                                // count=1 (valid user D#)
        g0[1] = lds_off;                                           // lds_addr
        g0[2] = (uint32_t)(ga & 0xFFFFFFFFu);                      // global_addr[31:0]
        g0[3] = (uint32_t)((ga >> 32) & 0x01FFFFFFu) | (2u << 30); // global_addr[56:32], type=2
        i32x8 g1;
        g1[0] = (int)(2u << 16);                                   // data_size=4B; no multicast/pad/iterate
        g1[1] = (int)(((uint32_t)L & 0xFFFFu) << 16);              // tensor_dim0[15:0]
        g1[2] = (int)((((uint32_t)L >> 16) & 0xFFFFu) | (1u << 16)); // tensor_dim0[31:16], tensor_dim1=1
        g1[3] = (int)(((uint32_t)L & 0xFFFFu) << 16);              // tensor_dim1 hi=0, tile_dim0=L
        g1[4] = 0;                                                 // tile_dim1=0, tile_dim2=0 (1-D tile)
        g1[5] = (int)L;                                            // tensor_dim0_stride
        g1[6] = 0;
        g1[7] = 0;
        i32x4 gz = {0, 0, 0, 0};                                   // groups 2/3 unused (<=2D)
#if defined(__clang_major__) && __clang_major__ >= 23
        // amdgpu-toolchain (clang-23 / therock-10.0): 6-arg form with extra i32x8
        i32x8 gz8 = {0, 0, 0, 0, 0, 0, 0, 0};
        __builtin_amdgcn_tensor_load_to_lds(g0, g1, gz, gz, gz8, 0);
#else
        // ROCm 7.2 (clang-22): 5-arg form
        __builtin_amdgcn_tensor_load_to_lds(g0, g1, gz, gz, 0);
#endif
        __builtin_amdgcn_s_wait_tensorcnt((short)0);
    }
#else
    for (int i = tid; i < L; i += 256) s_seqs[i] = seqs[(size_t)row * (size_t)L + i];
#endif
    __syncthreads();

    // ---- single streaming pass over the row ----
    float sum = 0.0f;   // sum(exp(x - predict))  (predict is a row element -> safe shift)
    int   cnt = 0;      // count(x > predict)

    const size_t base = (size_t)row * (size_t)V;
    int pre = (int)((4u - (uint32_t)(base & 3u)) & 3u);  // scalar prologue to 16B-align
    if (pre > V) pre = V;
    for (int i = tid; i < pre; i += 256) {
        float x = rowp[i];
        cnt += (x > predict);
        sum += fast_exp2(fmaf(x, LOG2E, negpk));
    }
    const int nvec = (V - pre) >> 2;
    const int tailstart = pre + (nvec << 2);
    const fx4* __restrict__ vp = (const fx4*)(rowp + pre);
    #pragma unroll 2
    for (int i = tid; i < nvec; i += 256) {
        fx4 v = NT_LOAD(vp + i);   // global_load_b128 th:NT (stream, read-once)
        cnt += (v.x > predict);
        sum += fast_exp2(fmaf(v.x, LOG2E, negpk));
        cnt += (v.y > predict);
        sum += fast_exp2(fmaf(v.y, LOG2E, negpk));
        cnt += (v.z > predict);
        sum += fast_exp2(fmaf(v.z, LOG2E, negpk));
        cnt += (v.w > predict);
        sum += fast_exp2(fmaf(v.w, LOG2E, negpk));
    }
    for (int i = tailstart + tid; i < V; i += 256) {
        float x = rowp[i];
        cnt += (x > predict);
        sum += fast_exp2(fmaf(x, LOG2E, negpk));
    }

    // ---- dedup'd seen-item correction (masked entries never exceed predict) ----
    if (tid < L) {
        int v = s_seqs[tid];
        bool first = true;
        for (int j = 0; j < tid; ++j)
            if (s_seqs[j] == v) { first = false; break; }
        if (first && rowp[v] > predict) atomicAdd(&s_seqcnt, 1);  // int: order-independent
    }

    // ---- deterministic reduction: wave32 shfl tree, then sequential combine ----
    #pragma unroll
    for (int off = 16; off >= 1; off >>= 1) {
        sum += __shfl_xor(sum, off, 32);
        cnt += __shfl_xor(cnt, off, 32);
    }
    const int wave = tid >> 5;
    if ((tid & 31) == 0) { s_sum[wave] = sum; s_cnt[wave] = cnt; }
    __syncthreads();

    if (tid == 0) {
        float S = 0.0f; int C = 0;
        #pragma unroll
        for (int w = 0; w < 8; ++w) { S += s_sum[w]; C += s_cnt[w]; }
        loss_out[row] = fast_log2(S) * LN2;               // -logp[label] = ln(sum exp(x - predict))
        rank_out[row] = (float)(C - s_seqcnt);
    }
}

// Fold B per-row (rank, loss) into the 10 metrics. One block, deterministic.
__global__ __launch_bounds__(256) void Ranker_reduce_kernel(
    const float* __restrict__ rank, const float* __restrict__ loss,
    float* __restrict__ out, int B)
{
    const int tid = threadIdx.x;
    float acc[10];
    #pragma unroll
    for (int j = 0; j < 10; ++j) acc[j] = 0.0f;

    const float ks[4] = {1.0f, 5.0f, 10.0f, 20.0f};
    for (int i = tid; i < B; i += 256) {
        float r = rank[i];
        acc[0] += loss[i];
        float invlog = 1.0f / fast_log2(r + 2.0f);   // 1/log2(rank+2)
        #pragma unroll
        for (int j = 0; j < 4; ++j) {
            float ind = (r < ks[j]) ? 1.0f : 0.0f;
            acc[1 + 2 * j] += ind * invlog;          // NDCG@k numerator
            acc[2 + 2 * j] += ind;                   // HR@k
        }
        acc[9] += 1.0f / (r + 1.0f);                 // MRR
    }

    #pragma unroll
    for (int j = 0; j < 10; ++j)
        #pragma unroll
        for (int off = 16; off >= 1; off >>= 1)
            acc[j] += __shfl_xor(acc[j], off, 32);

    __shared__ float s_acc[8][10];
    const int wave = tid >> 5;
    if ((tid & 31) == 0) {
        #pragma unroll
        for (int j = 0; j < 10; ++j) s_acc[wave][j] = acc[j];
    }
    __syncthreads();
    if (tid < 10) {
        float t = 0.0f;
        #pragma unroll
        for (int w = 0; w < 8; ++w) t += s_acc[w][tid];
        out[tid] = t / (float)B;
    }
}

extern "C" void kernel_launch(void* const* d_in, const int* in_sizes, int n_in,
                              void* d_out, int out_size, void* d_ws, size_t ws_size,
                              hipStream_t stream) {
    const float* scores = (const float*)d_in[0];
    const int*   labels = (const int*)d_in[1];
    const int*   seqs   = (const int*)d_in[2];
    const int B = in_sizes[1];
    const int V = in_sizes[0] / B;
    const int L = in_sizes[2] / B;

    float* rank = (float*)d_ws;        // B floats
    float* loss = rank + B;            // B floats

    Ranker_row_kernel<<<B, 256, 0, stream>>>(scores, labels, seqs, rank, loss, V, L);
    Ranker_reduce_kernel<<<1, 256, 0, stream>>>(rank, loss, (float*)d_out, B);
}